// HyperConnection_18794776887506
// MI455X (gfx1250) — compile-verified
//
#include <hip/hip_runtime.h>
#include <hip/hip_bf16.h>

typedef __attribute__((ext_vector_type(2))) float v2f;
typedef __attribute__((ext_vector_type(8))) float v8f;

#define HID 2048
#define NEXP 4
#define HC_EPS 1e-5f

// ---------------------------------------------------------------------------
// Prep kernel 1: pack W_beta[2048x4], W_m[2048x4], W_r[2048x16] into
// Wpack[2048][32] (cols 0-3 beta, 4-7 m, 8-23 r, 24-31 zero padding).
// ---------------------------------------------------------------------------
__global__ void hc_pack_w(const float* __restrict__ Wb, const float* __restrict__ Wm,
                          const float* __restrict__ Wr, float* __restrict__ Wpack) {
    int idx = blockIdx.x * 256 + threadIdx.x;   // 0 .. 2048*32-1
    int d = idx >> 5, c = idx & 31;
    float v = 0.f;
    if (c < 4)       v = Wb[d * 4 + c];
    else if (c < 8)  v = Wm[d * 4 + (c - 4)];
    else if (c < 24) v = Wr[d * 16 + (c - 8)];
    Wpack[idx] = v;
}

// ---------------------------------------------------------------------------
// Prep kernel 2: column sums of Wpack (needed to fold the LN mean into the
// raw-row GEMM results).  One block per column.
// ---------------------------------------------------------------------------
__global__ void hc_csum(const float* __restrict__ Wpack, float* __restrict__ csum) {
    __shared__ float red[256];
    int c = blockIdx.x;
    float s = 0.f;
    for (int d = threadIdx.x; d < HID; d += 256) s += Wpack[d * 32 + c];
    red[threadIdx.x] = s;
    __syncthreads();
    for (int o = 128; o > 0; o >>= 1) {
        if ((int)threadIdx.x < o) red[threadIdx.x] += red[threadIdx.x + o];
        __syncthreads();
    }
    if (threadIdx.x == 0) csum[c] = red[0];
}

// ---------------------------------------------------------------------------
// Main fused kernel: 4 tokens per block, 256 threads (8 wave32).
// ---------------------------------------------------------------------------
__launch_bounds__(256)
__global__ void hc_main(const float* __restrict__ lo_g,    // [8192, 2048]
                        const float* __restrict__ h_g,     // [8192, 4, 2048]
                        const float* __restrict__ Bvec,    // [1,4]
                        const float* __restrict__ Amvec,   // [4,1]
                        const float* __restrict__ Arm,     // [4,4]
                        const float* __restrict__ s_alpha, // [4,4]
                        const float* __restrict__ s_beta,  // [1,4]
                        const float* __restrict__ Wpack,   // [2048,32]
                        const float* __restrict__ csum,    // [32]
                        float* __restrict__ out)           // [8192,4,2048]
{
    __shared__ float h_lds[16 * HID];       // 128 KB: 4 tokens x 4 rows raw
    __shared__ float P_part[8][16 * 32];    // 16 KB: per-wave partial tiles
    __shared__ float P_lds[16 * 32];        // reduced row-projections
    __shared__ float stat1[16];
    __shared__ float stat2[16];
    __shared__ float M_lds[4][4][4];        // Ar_dyn + Am_dyn broadcast
    __shared__ float Bd_lds[4][4];

    const int tid  = threadIdx.x;
    const int wave = tid >> 5;
    const int lane = tid & 31;
    const long token0 = (long)blockIdx.x * 4;

    // ---- Stage 1: stream 16 rows into LDS, per-row sum / sumsq -------------
    {
        const float4* src = (const float4*)(h_g + token0 * (NEXP * HID));
        float4* dst = (float4*)h_lds;
        #pragma unroll
        for (int rr = 0; rr < 2; ++rr) {
            const int row = wave * 2 + rr;          // 8 waves x 2 rows = 16
            float s1 = 0.f, s2 = 0.f;
            #pragma unroll
            for (int k = 0; k < 16; ++k) {
                int f4 = row * 512 + lane + 32 * k;
                float4 v = src[f4];
                dst[f4] = v;
                s1 += v.x + v.y + v.z + v.w;
                s2 += v.x * v.x + v.y * v.y + v.z * v.z + v.w * v.w;
            }
            #pragma unroll
            for (int off = 16; off >= 1; off >>= 1) {
                s1 += __shfl_xor(s1, off, 32);
                s2 += __shfl_xor(s2, off, 32);
            }
            if (lane == 0) { stat1[row] = s1; stat2[row] = s2; }
        }
    }
    __syncthreads();

    // ---- Stage 2: P[16x32] = raw rows @ Wpack via V_WMMA_F32_16X16X4_F32 ---
    {
        const int m    = lane & 15;    // A row / B-C column within tile
        const int half = lane >> 4;    // K-half select per ISA layout
        const int kbase = wave * 256;  // each wave owns 256 of K=2048
        const float* hrow = h_lds + m * HID;
        v8f acc0 = {0.f, 0.f, 0.f, 0.f, 0.f, 0.f, 0.f, 0.f};
        v8f acc1 = acc0;
        for (int kk = 0; kk < 64; ++kk) {
            int k = kbase + kk * 4 + half * 2;
            v2f A;  A.x  = hrow[k];                A.y  = hrow[k + 1];
            v2f B0; B0.x = Wpack[k * 32 + m];      B0.y = Wpack[(k + 1) * 32 + m];
            v2f B1; B1.x = Wpack[k * 32 + 16 + m]; B1.y = Wpack[(k + 1) * 32 + 16 + m];
            acc0 = __builtin_amdgcn_wmma_f32_16x16x4_f32(false, A, false, B0,
                                                         (short)0, acc0, false, false);
            acc1 = __builtin_amdgcn_wmma_f32_16x16x4_f32(false, A, false, B1,
                                                         (short)0, acc1, false, false);
        }
        // C layout: vgpr v, lane l -> row = v + 8*(l>>4), col = l&15
        #pragma unroll
        for (int v = 0; v < 8; ++v) {
            int row = v + 8 * half;
            P_part[wave][row * 32 + m]      = acc0[v];
            P_part[wave][row * 32 + 16 + m] = acc1[v];
        }
    }
    __syncthreads();

    // deterministic fixed-order cross-wave tile reduction
    for (int o = tid; o < 16 * 32; o += 256) {
        float s = 0.f;
        #pragma unroll
        for (int w = 0; w < 8; ++w) s += P_part[w][o];
        P_lds[o] = s;
    }
    __syncthreads();

    // ---- Stage 3: dynamic coefficients per token ---------------------------
    if (tid < 64) {
        const int t = tid >> 4, i = (tid >> 2) & 3, j = tid & 3;
        float mu[4], inv[4];
        #pragma unroll
        for (int n = 0; n < 4; ++n) {
            float m1 = stat1[4 * t + n] * (1.f / HID);
            float m2 = stat2[4 * t + n] * (1.f / HID);
            mu[n]  = m1;
            inv[n] = rsqrtf(m2 - m1 * m1 + HC_EPS);
        }
        // H_bar @ Wpack[:,c] = 0.25 * sum_n inv_n * (P[row_n,c] - mu_n*csum[c])
        auto proj = [&](int c) {
            float cs = csum[c];
            float s = 0.f;
            #pragma unroll
            for (int n = 0; n < 4; ++n)
                s += inv[n] * (P_lds[(4 * t + n) * 32 + c] - mu[n] * cs);
            return 0.25f * s;
        };
        float ar = s_alpha[i * 4 + j] * tanhf(proj(8 + 4 * i + j)) + Arm[i * 4 + j];
        float am = s_alpha[j * 4]     * tanhf(proj(4 + j))         + Amvec[j];
        M_lds[t][i][j] = ar + am;                 // residual + broadcast mix
        if (i == 0)
            Bd_lds[t][j] = s_beta[j] * tanhf(proj(j)) + Bvec[j];
    }
    __syncthreads();

    // ---- Stage 4: out[i,d] = Bd[i]*lo[d] + sum_j M[i][j]*h[j,d] ------------
    {
        const float4* h4 = (const float4*)h_lds;
        const float4* lo4g = (const float4*)lo_g;
        float4* out4 = (float4*)out;
        #pragma unroll
        for (int t = 0; t < 4; ++t) {
            const long tok = token0 + t;
            #pragma unroll
            for (int k2 = 0; k2 < 2; ++k2) {
                const int d4 = tid + 256 * k2;          // 512 float4 per row
                float4 lo = lo4g[tok * 512 + d4];
                float4 hj[4];
                #pragma unroll
                for (int j = 0; j < 4; ++j) hj[j] = h4[(4 * t + j) * 512 + d4];
                #pragma unroll
                for (int i = 0; i < 4; ++i) {
                    float bd = Bd_lds[t][i];
                    float4 o;
                    o.x = bd * lo.x; o.y = bd * lo.y;
                    o.z = bd * lo.z; o.w = bd * lo.w;
                    #pragma unroll
                    for (int j = 0; j < 4; ++j) {
                        float mm = M_lds[t][i][j];
                        o.x += mm * hj[j].x; o.y += mm * hj[j].y;
                        o.z += mm * hj[j].z; o.w += mm * hj[j].w;
                    }
                    out4[tok * (NEXP * 512) + i * 512 + d4] = o;
                }
            }
        }
    }
}

extern "C" void kernel_launch(void* const* d_in, const int* in_sizes, int n_in,
                              void* d_out, int out_size, void* d_ws, size_t ws_size,
                              hipStream_t stream) {
    const float* lo = (const float*)d_in[0];   // layer_output [4,2048,2048]
    const float* hs = (const float*)d_in[1];   // hidden_states [4,2048,4,2048]
    const float* B  = (const float*)d_in[2];
    const float* Am = (const float*)d_in[3];
    const float* Ar = (const float*)d_in[4];
    const float* sa = (const float*)d_in[5];
    const float* sb = (const float*)d_in[6];
    const float* Wb = (const float*)d_in[7];
    const float* Wm = (const float*)d_in[8];
    const float* Wr = (const float*)d_in[9];
    float* out = (float*)d_out;

    float* Wpack = (float*)d_ws;               // 2048*32 floats
    float* csum  = Wpack + HID * 32;           // 32 floats

    const int tokens = in_sizes[0] / HID;      // 8192
    hc_pack_w<<<(HID * 32) / 256, 256, 0, stream>>>(Wb, Wm, Wr, Wpack);
    hc_csum<<<32, 256, 0, stream>>>(Wpack, csum);
    hc_main<<<tokens / 4, 256, 0, stream>>>(lo, hs, B, Am, Ar, sa, sb,
                                            Wpack, csum, out);
}